// GINModel_67516885893485
// MI455X (gfx1250) — compile-verified
//
#include <hip/hip_runtime.h>
#include <hip/hip_bf16.h>

typedef __attribute__((ext_vector_type(2))) float v2f;
typedef __attribute__((ext_vector_type(8))) float v8f;

#define C_CH 128   // channels
#define NT 4       // 16-col tiles per wave in GEMM (wave computes 16x64)

// ---------------------------------------------------------------------------
// h[n,:] = x[n,:] @ ntW + ntb   (x is [N,2], ntW [2,128]); pad rows -> 0
// ---------------------------------------------------------------------------
__global__ void node_transform_kernel(const float* __restrict__ x,
                                      const float* __restrict__ ntW,
                                      const float* __restrict__ ntb,
                                      float* __restrict__ h,
                                      int N, int Mpad) {
    long t = (long)blockIdx.x * blockDim.x + threadIdx.x;
    long n = t >> 5;
    int c = (int)(t & 31) * 4;
    if (n >= Mpad) return;
    float4 o;
    if (n < N) {
        float x0 = x[n * 2 + 0];
        float x1 = x[n * 2 + 1];
        o.x = x0 * ntW[c + 0] + x1 * ntW[C_CH + c + 0] + ntb[c + 0];
        o.y = x0 * ntW[c + 1] + x1 * ntW[C_CH + c + 1] + ntb[c + 1];
        o.z = x0 * ntW[c + 2] + x1 * ntW[C_CH + c + 2] + ntb[c + 2];
        o.w = x0 * ntW[c + 3] + x1 * ntW[C_CH + c + 3] + ntb[c + 3];
    } else {
        o = make_float4(0.f, 0.f, 0.f, 0.f);
    }
    *(float4*)(h + n * C_CH + c) = o;
}

// ---------------------------------------------------------------------------
// z[dst[e],:] += h[src[e],:]  (z pre-initialized with h => z = h + agg)
// one wave per edge (src/dst uniform across wave), float4 per lane;
// agg buffer (51 MB) is L2-resident on MI455X (192 MB L2).
// ---------------------------------------------------------------------------
__global__ void scatter_add_kernel(const float4* __restrict__ h4,
                                   const int* __restrict__ src,
                                   const int* __restrict__ dst,
                                   float* __restrict__ z, int E) {
    long t = (long)blockIdx.x * blockDim.x + threadIdx.x;
    long e = t >> 5;
    int c4 = (int)(t & 31);
    if (e >= E) return;
    int s = src[e];
    int d = dst[e];
    float4 v = h4[(long)s * (C_CH / 4) + c4];
    float* p = z + (long)d * C_CH + c4 * 4;
    atomicAdd(p + 0, v.x);
    atomicAdd(p + 1, v.y);
    atomicAdd(p + 2, v.z);
    atomicAdd(p + 3, v.w);
}

// ---------------------------------------------------------------------------
// out[m,n] = relu( (sum_k A[m,k]*W[k,n]) * scale[n] + biasEff[n] )
//   scale==null -> scale=1, biasEff = b1[n]
//   scale!=null -> biasEff = b1[n]*scale[n] + b2[n]   (BN affine folded)
//
// fp32 WMMA 16x16x4. Block = 8 waves, tile = 128 rows x 64 cols,
// each wave: 16 rows x 4x16 cols (4 accumulators reuse one A fragment).
//
// Weight panel is staged in LDS in *fragment-ready* order so that each
// lane's B operand pair {K=k+2*half, K=k+2*half+1} is one contiguous b64:
//   ldsB[ ((s*NT + t)*32 + lane)*2 + p ] = Wscaled[4s + 2*(lane>>4) + p,
//                                                  n0 + 16t + (lane&15)]
// Inner-loop B fetch = single ds_load_b64, 32 lanes cover one full 256B
// LDS bank row -> conflict-free, no repack movs before v_wmma.
// Pad rows of A are zero -> EXEC all-1s, no predication around WMMA.
// ---------------------------------------------------------------------------
__global__ void gemm_bn_relu_wmma_kernel(const float* __restrict__ A,
                                         const float* __restrict__ W,
                                         const float* __restrict__ scale,
                                         const float* __restrict__ b1,
                                         const float* __restrict__ b2,
                                         float* __restrict__ out,
                                         int K, int NCOL) {
    extern __shared__ float ldsB[];   // K/4 steps * NT tiles * 32 lanes * 2
    const int tid = threadIdx.x;
    const int n0 = blockIdx.y * (16 * NT);

    // stage scaled weights, fragment-ready layout
    for (int i = tid; i < K * (16 * NT); i += 256) {
        int p  = i & 1;
        int ln = (i >> 1) & 31;
        int t  = (i >> 6) & (NT - 1);
        int s  = i >> 8;                       // i / (64*NT/... ) == step
        int kk = 4 * s + 2 * (ln >> 4) + p;
        int nn = 16 * t + (ln & 15);
        float w = W[(long)kk * NCOL + n0 + nn];
        if (scale) w *= scale[n0 + nn];
        ldsB[i] = w;
    }
    __syncthreads();

    const int wave = tid >> 5;     // 0..7
    const int lane = tid & 31;
    const int half = lane >> 4;    // 0: lanes 0-15, 1: lanes 16-31
    const int l15  = lane & 15;
    const long m0 = (long)blockIdx.x * 128 + wave * 16;

    // A fragment: lane covers row (m0+l15), K pair {k+2*half, k+2*half+1}
    const float* Arow = A + (m0 + l15) * K + half * 2;
    const v2f* Bfrag = (const v2f*)ldsB + lane;   // + (s*NT + t)*32

    v8f acc[NT];
    #pragma unroll
    for (int t = 0; t < NT; ++t) acc[t] = (v8f){};

    const int steps = K >> 2;
    for (int s = 0; s < steps; ++s) {
        v2f a;
        a.x = Arow[4 * s];          // contiguous pair -> global_load_b64
        a.y = Arow[4 * s + 1];
        #pragma unroll
        for (int t = 0; t < NT; ++t) {
            v2f b = Bfrag[(s * NT + t) * 32];   // single ds_load_b64
            acc[t] = __builtin_amdgcn_wmma_f32_16x16x4_f32(
                false, a, false, b, (short)0, acc[t], false, false);
        }
    }

    // epilogue: fused bias (+BN affine) + ReLU; C/D layout per ISA 7.12.2
    #pragma unroll
    for (int t = 0; t < NT; ++t) {
        int n = n0 + t * 16 + l15;
        float bias = b1 ? b1[n] : 0.f;
        if (scale) bias = bias * scale[n] + (b2 ? b2[n] : 0.f);
        float* op = out + (m0 + half * 8) * NCOL + n;
        #pragma unroll
        for (int r = 0; r < 8; ++r) {
            float v = acc[t][r] + bias;
            op[(long)r * NCOL] = v > 0.f ? v : 0.f;
        }
    }
}

// ---------------------------------------------------------------------------
// g[batch[n],:] += h[n,:]   (global sum pool; g pre-zeroed)
// ---------------------------------------------------------------------------
__global__ void pool_kernel(const float4* __restrict__ h4,
                            const int* __restrict__ batch,
                            float* __restrict__ g, int N) {
    long t = (long)blockIdx.x * blockDim.x + threadIdx.x;
    long n = t >> 5;
    int c4 = (int)(t & 31);
    if (n >= N) return;
    int b = batch[n];
    float4 v = h4[n * (C_CH / 4) + c4];
    float* p = g + (long)b * C_CH + c4 * 4;
    atomicAdd(p + 0, v.x);
    atomicAdd(p + 1, v.y);
    atomicAdd(p + 2, v.z);
    atomicAdd(p + 3, v.w);
}

// ---------------------------------------------------------------------------
// y = elu(g @ pW1 + pb1) @ pW2 + pb2 ; out = log_softmax(y)
// one block, 256 threads; elu intermediate [128 x 64] in LDS
// ---------------------------------------------------------------------------
__global__ void head_kernel(const float* __restrict__ g,
                            const float* __restrict__ pW1,
                            const float* __restrict__ pb1,
                            const float* __restrict__ pW2,
                            const float* __restrict__ pb2,
                            float* __restrict__ out, int G) {
    __shared__ float e[128 * 64];
    const int tid = threadIdx.x;
    for (int i = tid; i < G * 64; i += 256) {
        int gi = i >> 6;
        int j  = i & 63;
        float acc = pb1[j];
        const float* gr = g + (long)gi * C_CH;
        #pragma unroll 8
        for (int k = 0; k < C_CH; ++k) acc += gr[k] * pW1[k * 64 + j];
        e[i] = acc > 0.f ? acc : (__expf(acc) - 1.f);
    }
    __syncthreads();
    if (tid < G) {
        const float* er = e + tid * 64;
        float y0 = pb2[0], y1 = pb2[1];
        #pragma unroll 8
        for (int j = 0; j < 64; ++j) {
            y0 += er[j] * pW2[2 * j + 0];
            y1 += er[j] * pW2[2 * j + 1];
        }
        float m  = fmaxf(y0, y1);
        float ls = m + logf(expf(y0 - m) + expf(y1 - m));
        out[tid * 2 + 0] = y0 - ls;
        out[tid * 2 + 1] = y1 - ls;
    }
}

// ---------------------------------------------------------------------------
extern "C" void kernel_launch(void* const* d_in, const int* in_sizes, int n_in,
                              void* d_out, int out_size, void* d_ws, size_t ws_size,
                              hipStream_t stream) {
    const float* x     = (const float*)d_in[0];
    const int*   ei    = (const int*)  d_in[1];
    const int*   batch = (const int*)  d_in[2];
    const float* ntW   = (const float*)d_in[3];
    const float* ntb   = (const float*)d_in[4];
    const float* W1    = (const float*)d_in[5];
    const float* b1    = (const float*)d_in[6];
    const float* g1    = (const float*)d_in[7];
    const float* be1   = (const float*)d_in[8];
    const float* W2    = (const float*)d_in[9];
    const float* b2    = (const float*)d_in[10];
    const float* pW1   = (const float*)d_in[11];
    const float* pb1   = (const float*)d_in[12];
    const float* pW2   = (const float*)d_in[13];
    const float* pb2   = (const float*)d_in[14];
    float* out = (float*)d_out;

    const int N = in_sizes[0] / 2;          // 100000
    const int E = in_sizes[1] / 2;          // 3200000
    const int G = out_size / 2;             // 128
    const int C = C_CH;                     // 128
    const int C2 = 2 * C;                   // 256
    const int L = in_sizes[5] / (C * C2);   // 4
    const int Mpad = ((N + 127) / 128) * 128;

    const int* src = ei;
    const int* dst = ei + E;

    // workspace carve-out
    char* ws = (char*)d_ws;
    float* h  = (float*)ws;                        ws += (size_t)Mpad * C  * sizeof(float);
    float* z  = (float*)ws;                        ws += (size_t)Mpad * C  * sizeof(float);
    float* z2 = (float*)ws;                        ws += (size_t)Mpad * C2 * sizeof(float);
    float* gp = (float*)ws;                        ws += (size_t)G    * C  * sizeof(float);
    (void)ws_size;

    const int TB = 256;

    // 1. node transform (zeroes pad rows so WMMA needs no predication)
    {
        long thr = (long)Mpad * 32;
        node_transform_kernel<<<(int)((thr + TB - 1) / TB), TB, 0, stream>>>(
            x, ntW, ntb, h, N, Mpad);
    }

    // 2. GIN layers
    for (int l = 0; l < L; ++l) {
        // z = h  (eps=0: z starts at h, scatter adds neighbors)
        hipMemcpyAsync(z, h, (size_t)Mpad * C * sizeof(float),
                       hipMemcpyDeviceToDevice, stream);
        {
            long thr = (long)E * 32;
            scatter_add_kernel<<<(int)((thr + TB - 1) / TB), TB, 0, stream>>>(
                (const float4*)h, src, dst, z, E);
        }
        // z2 = relu(BN(z @ W1[l] + b1[l]))   [Mpad, 256]
        {
            dim3 grid(Mpad / 128, C2 / (16 * NT));
            size_t smem = (size_t)C * (16 * NT) * sizeof(float);   // 32 KB
            gemm_bn_relu_wmma_kernel<<<grid, TB, smem, stream>>>(
                z, W1 + (size_t)l * C * C2, g1 + (size_t)l * C2,
                b1 + (size_t)l * C2, be1 + (size_t)l * C2, z2, C, C2);
        }
        // h = relu(z2 @ W2[l] + b2[l])       [Mpad, 128]
        {
            dim3 grid(Mpad / 128, C / (16 * NT));
            size_t smem = (size_t)C2 * (16 * NT) * sizeof(float);  // 64 KB
            gemm_bn_relu_wmma_kernel<<<grid, TB, smem, stream>>>(
                z2, W2 + (size_t)l * C2 * C, nullptr,
                b2 + (size_t)l * C, nullptr, h, C2, C);
        }
    }

    // 3. global sum pool
    hipMemsetAsync(gp, 0, (size_t)G * C * sizeof(float), stream);
    {
        long thr = (long)N * 32;
        pool_kernel<<<(int)((thr + TB - 1) / TB), TB, 0, stream>>>(
            (const float4*)h, batch, gp, N);
    }

    // 4. MLP head + log_softmax
    head_kernel<<<1, TB, 0, stream>>>(gp, pW1, pb1, pW2, pb2, out, G);
}